// EditDistNeuralModelProgressive_40046275068393
// MI455X (gfx1250) — compile-verified
//
#include <hip/hip_runtime.h>

#define NEGINF (-__builtin_inff())

constexpr int Bb = 64, Tt = 256, Vv = 256, Cc = 4;
constexpr int ROWE = Vv * Cc;                 // 1024 floats per (b,t) score row
constexpr int DQ   = 4;                       // TDM prefetch depth (ring)

// ---------------------------------------------------------------------------
// Tensor Data Mover: DMA one contiguous 1024-float score row (4 KB) into LDS.
// D# layout per CDNA5 ISA ch.7/8 (group0: count|lds_addr|global_addr|type=2,
// group1: data_size=4B, tensor_dim0 = tile_dim0 = 1024, 1 row).
// ---------------------------------------------------------------------------
typedef __attribute__((ext_vector_type(4))) unsigned int u32x4;
typedef __attribute__((ext_vector_type(4))) int          i32x4;
typedef __attribute__((ext_vector_type(8))) int          i32x8;

__device__ __forceinline__ void tdm_load_row(const void* gsrc, unsigned lds_addr) {
  unsigned long long ga = (unsigned long long)gsrc;
  u32x4 g0 = { 1u,                                   // count = 1 valid descriptor
               lds_addr,                             // LDS byte address
               (unsigned)ga,                         // global_addr[31:0]
               (unsigned)((ga >> 32) & 0x01FFFFFFull) | (2u << 30) }; // [56:32] | type=2
  const unsigned n = (unsigned)ROWE;                 // 1024 elements of 4 bytes
  i32x8 g1 = { (int)(2u << 16),                      // data_size = 2 (4 bytes)
               (int)((n & 0xFFFFu) << 16),           // tensor_dim0[15:0]
               (int)((n >> 16) | (1u << 16)),        // tensor_dim0[31:16] | tensor_dim1=1
               (int)(n << 16),                       // tile_dim0 = 1024
               1,                                    // tile_dim1 = 1, tile_dim2 = 0
               (int)n,                               // tensor_dim0_stride
               0, 0 };
  i32x4 z4 = { 0, 0, 0, 0 };
#if __clang_major__ >= 23
  i32x8 z8 = { 0, 0, 0, 0, 0, 0, 0, 0 };
  __builtin_amdgcn_tensor_load_to_lds(g0, g1, z4, z4, z8, 0);
#else
  __builtin_amdgcn_tensor_load_to_lds(g0, g1, z4, z4, 0);
#endif
}

// Wait until <= DQ-1 TDM loads outstanding: with exactly DQ in flight and
// in-order completion, this guarantees the oldest row is resident in LDS.
__device__ __forceinline__ void tdm_wait_ring() {
  __builtin_amdgcn_s_wait_tensorcnt(DQ - 1);
}

// ---------------------------------------------------------------------------
// Numerics
// ---------------------------------------------------------------------------
__device__ __forceinline__ float laexp(float a, float b) {   // jnp.logaddexp
  float m = fmaxf(a, b), n = fminf(a, b);
  if (m == NEGINF) return NEGINF;
  return m + log1pf(expf(n - m));
}

// Inclusive +scan over 256 elems (8 regs/lane, wave32 shuffle). No barriers.
__device__ __forceinline__ void add_scan8_fwd(float x[8]) {
  const int lane = threadIdx.x;
  #pragma unroll
  for (int j = 1; j < 8; ++j) x[j] += x[j - 1];
  float inc = x[7];
  #pragma unroll
  for (int off = 1; off < 32; off <<= 1) {
    float y = __shfl_up(inc, off, 32);
    if (lane >= off) inc += y;
  }
  float ex = __shfl_up(inc, 1, 32);
  ex = (lane == 0) ? 0.f : ex;
  #pragma unroll
  for (int j = 0; j < 8; ++j) x[j] += ex;
}

__device__ __forceinline__ void add_scan8_rev(float x[8]) {
  const int lane = threadIdx.x;
  #pragma unroll
  for (int j = 6; j >= 0; --j) x[j] += x[j + 1];
  float inc = x[0];
  #pragma unroll
  for (int off = 1; off < 32; off <<= 1) {
    float y = __shfl_down(inc, off, 32);
    if (lane + off < 32) inc += y;
  }
  float ex = __shfl_down(inc, 1, 32);
  ex = (lane == 31) ? 0.f : ex;
  #pragma unroll
  for (int j = 0; j < 8; ++j) x[j] += ex;
}

// Cumulative logsumexp via (max, scaled-sum) pairs: one trans-op of latency
// per combine (the two exps are independent), single log at the end.
// Inputs are finite; identity (m=-inf, s=0) only ever combines with finite m.
__device__ __forceinline__ void lse_scan8_fwd(float x[8]) {
  const int lane = threadIdx.x;
  float m[8], s[8];
  m[0] = x[0]; s[0] = 1.f;
  #pragma unroll
  for (int j = 1; j < 8; ++j) {
    float M = fmaxf(m[j - 1], x[j]);
    s[j] = s[j - 1] * expf(m[j - 1] - M) + expf(x[j] - M);
    m[j] = M;
  }
  float wm = m[7], ws = s[7];
  #pragma unroll
  for (int off = 1; off < 32; off <<= 1) {
    float om = __shfl_up(wm, off, 32);
    float os = __shfl_up(ws, off, 32);
    if (lane >= off) {
      float M = fmaxf(wm, om);
      ws = ws * expf(wm - M) + os * expf(om - M);
      wm = M;
    }
  }
  float exm = __shfl_up(wm, 1, 32);
  float exs = __shfl_up(ws, 1, 32);
  exm = (lane == 0) ? NEGINF : exm;            // identity prefix for lane 0
  exs = (lane == 0) ? 0.f    : exs;
  #pragma unroll
  for (int j = 0; j < 8; ++j) {
    float M = fmaxf(m[j], exm);
    float S = s[j] * expf(m[j] - M) + exs * expf(exm - M);
    x[j] = M + logf(S);
  }
}

__device__ __forceinline__ void lse_scan8_rev(float x[8]) {
  const int lane = threadIdx.x;
  float m[8], s[8];
  m[7] = x[7]; s[7] = 1.f;
  #pragma unroll
  for (int j = 6; j >= 0; --j) {
    float M = fmaxf(m[j + 1], x[j]);
    s[j] = s[j + 1] * expf(m[j + 1] - M) + expf(x[j] - M);
    m[j] = M;
  }
  float wm = m[0], ws = s[0];
  #pragma unroll
  for (int off = 1; off < 32; off <<= 1) {
    float om = __shfl_down(wm, off, 32);
    float os = __shfl_down(ws, off, 32);
    if (lane + off < 32) {
      float M = fmaxf(wm, om);
      ws = ws * expf(wm - M) + os * expf(om - M);
      wm = M;
    }
  }
  float exm = __shfl_down(wm, 1, 32);
  float exs = __shfl_down(ws, 1, 32);
  exm = (lane == 31) ? NEGINF : exm;
  exs = (lane == 31) ? 0.f    : exs;
  #pragma unroll
  for (int j = 0; j < 8; ++j) {
    float M = fmaxf(m[j], exm);
    float S = s[j] * expf(m[j] - M) + exs * expf(exm - M);
    x[j] = M + logf(S);
  }
}

__device__ __forceinline__ void load_row(const float* rb, int lane,
                                         float del[8], float ins[8], float sub[8]) {
  const float4* r4 = (const float4*)rb + lane * 8;   // ds_load_b128 x8
  #pragma unroll
  for (int j = 0; j < 8; ++j) {
    float4 q = r4[j];
    del[j] = q.x; ins[j] = q.y; sub[j] = q.z;
  }
}

__device__ __forceinline__ void store_row(float* dst, int lane, const float x[8]) {
  float4* d4 = (float4*)dst + lane * 2;
  d4[0] = make_float4(x[0], x[1], x[2], x[3]);
  d4[1] = make_float4(x[4], x[5], x[6], x[7]);
}

// ---------------------------------------------------------------------------
// Kernel 1: log_softmax over the class dim (pure streaming, float4/b128)
// ---------------------------------------------------------------------------
__global__ __launch_bounds__(256) void k_logsoftmax(const float4* __restrict__ in,
                                                    float4* __restrict__ out, int n4) {
  int i = blockIdx.x * blockDim.x + threadIdx.x;
  if (i >= n4) return;
  float4 x = in[i];
  float m = fmaxf(fmaxf(x.x, x.y), fmaxf(x.z, x.w));
  float l = m + logf(expf(x.x - m) + expf(x.y - m) + expf(x.z - m) + expf(x.w - m));
  out[i] = make_float4(x.x - l, x.y - l, x.z - l, x.w - l);
}

// ---------------------------------------------------------------------------
// Kernel 2: alpha (blocks 0..B-1) / beta (blocks B..2B-1). One wave per block.
// 4-deep TDM ring hides HBM/L2 latency behind the row scans; dummy loads keep
// exactly DQ transfers in flight so s_wait_tensorcnt(DQ-1) is always correct.
// ---------------------------------------------------------------------------
__global__ __launch_bounds__(32) void k_dp(const float* __restrict__ scores,
                                           float* __restrict__ alpha,
                                           float* __restrict__ beta,
                                           float* __restrict__ scalar_out) {
  __shared__ __align__(16) float rowbuf[DQ + 1][ROWE];  // 4 ring slots + 1 dummy = 20 KB
  const int  lane    = threadIdx.x;
  const bool is_beta = (blockIdx.x >= (unsigned)Bb);
  const int  b       = is_beta ? (int)blockIdx.x - Bb : (int)blockIdx.x;
  const float* base  = scores + (size_t)b * Tt * ROWE;
  const unsigned lds_base = (unsigned)(size_t)(void*)&rowbuf[0][0];
  auto lds_of = [&](int slot) { return lds_base + (unsigned)slot * (ROWE * 4); };
  float del[8], ins[8], sub[8];

  if (!is_beta) {
    float* aw = alpha + (size_t)b * Tt * Vv;
    #pragma unroll
    for (int r = 0; r < DQ; ++r)                       // prologue: rows 0..3
      tdm_load_row(base + (size_t)r * ROWE, lds_of(r));
    tdm_wait_ring();                                   // row 0 resident
    load_row(&rowbuf[0][0], lane, del, ins, sub);
    float prev[8];
    #pragma unroll
    for (int j = 0; j < 8; ++j) prev[j] = (lane == 0 && j == 0) ? 0.f : ins[j];
    asm volatile("s_wait_dscnt 0" ::: "memory");       // slot 0 reads done
    tdm_load_row(base + (size_t)DQ * ROWE, lds_of(0)); // refill slot 0 with row 4
    add_scan8_fwd(prev);                               // alpha0 = [0, cumsum(ins0[1:])]
    store_row(aw, lane, prev);
    for (int t = 1; t < Tt; ++t) {
      tdm_wait_ring();                                 // row t resident
      load_row(&rowbuf[t & (DQ - 1)][0], lane, del, ins, sub);
      float pm1 = __shfl_up(prev[7], 1, 32);           // prev[v-1] across lanes
      float c[8];
      c[0] = (lane == 0) ? (del[0] + prev[0]) : laexp(del[0] + prev[0], sub[0] + pm1);
      #pragma unroll
      for (int j = 1; j < 8; ++j) c[j] = laexp(del[j] + prev[j], sub[j] + prev[j - 1]);
      float s[8];
      #pragma unroll
      for (int j = 0; j < 8; ++j) s[j] = (lane == 0 && j == 0) ? 0.f : ins[j];
      asm volatile("s_wait_dscnt 0" ::: "memory");     // slot reads done before DMA refill
      int nt = t + DQ;
      if (nt < Tt) tdm_load_row(base + (size_t)nt * ROWE, lds_of(t & (DQ - 1)));
      else         tdm_load_row(base + (size_t)(Tt - 1) * ROWE, lds_of(DQ));  // dummy
      add_scan8_fwd(s);                                // s = [0, cumsum(ins[1:])]
      float y[8];
      #pragma unroll
      for (int j = 0; j < 8; ++j) y[j] = c[j] - s[j];
      lse_scan8_fwd(y);                                // cumulative logsumexp
      #pragma unroll
      for (int j = 0; j < 8; ++j) prev[j] = s[j] + y[j];
      store_row(aw + (size_t)t * Vv, lane, prev);
    }
    if (b == 0 && lane == 31) scalar_out[0] = prev[7]; // alpha[0, T-1, V-1]
  } else {
    float* bw = beta + (size_t)b * Tt * Vv;
    #pragma unroll
    for (int r = 0; r < DQ; ++r) {                     // prologue: rows 255..252
      int row = Tt - 1 - r;
      tdm_load_row(base + (size_t)row * ROWE, lds_of(row & (DQ - 1)));
    }
    tdm_wait_ring();                                   // row T-1 resident (slot 3)
    load_row(&rowbuf[(Tt - 1) & (DQ - 1)][0], lane, del, ins, sub);
    float nxt[8];
    #pragma unroll
    for (int j = 0; j < 8; ++j) nxt[j] = (lane == 31 && j == 7) ? 0.f : ins[j];
    asm volatile("s_wait_dscnt 0" ::: "memory");
    tdm_load_row(base + (size_t)(Tt - 1 - DQ) * ROWE,
                 lds_of((Tt - 1 - DQ) & (DQ - 1)));    // row 251 -> slot 3
    add_scan8_rev(nxt);                                // beta_last = suffix-sum(ins[T-1])
    store_row(bw + (size_t)(Tt - 1) * Vv, lane, nxt);
    for (int t = Tt - 2; t >= 0; --t) {
      tdm_wait_ring();                                 // row t resident
      load_row(&rowbuf[t & (DQ - 1)][0], lane, del, ins, sub);
      float np1 = __shfl_down(nxt[0], 1, 32);          // nxt[v+1] across lanes
      float c[8];
      #pragma unroll
      for (int j = 0; j < 7; ++j) c[j] = laexp(del[j] + nxt[j], sub[j] + nxt[j + 1]);
      c[7] = (lane == 31) ? (del[7] + nxt[7]) : laexp(del[7] + nxt[7], sub[7] + np1);
      float S[8];
      #pragma unroll
      for (int j = 0; j < 8; ++j) S[j] = (lane == 31 && j == 7) ? 0.f : ins[j];
      asm volatile("s_wait_dscnt 0" ::: "memory");
      int nt = t - DQ;
      if (nt >= 0) tdm_load_row(base + (size_t)nt * ROWE, lds_of(t & (DQ - 1)));
      else         tdm_load_row(base, lds_of(DQ));     // dummy: row 0 -> spare slot
      add_scan8_rev(S);                                // suffix-sum of ins (excl. last)
      float y[8];
      #pragma unroll
      for (int j = 0; j < 8; ++j) y[j] = c[j] - S[j];
      lse_scan8_rev(y);                                // reverse cumulative logsumexp
      #pragma unroll
      for (int j = 0; j < 8; ++j) nxt[j] = S[j] + y[j];
      store_row(bw + (size_t)t * Vv, lane, nxt);
    }
  }
}

// ---------------------------------------------------------------------------
// Kernel 3: expected counts. Per class only one arc type is live, so
// ec = {ed, ei, es, -inf}; posterior = exp(ec - logsumexp(ec)).
// ---------------------------------------------------------------------------
__global__ __launch_bounds__(256) void k_expected(const float4* __restrict__ sc4,
                                                  const float* __restrict__ alpha,
                                                  const float* __restrict__ beta,
                                                  float4* __restrict__ out4) {
  int i = blockIdx.x * blockDim.x + threadIdx.x;
  if (i >= Bb * Tt * Vv) return;
  int v = i & (Vv - 1);
  int t = (i >> 8) & (Tt - 1);
  float4 s = sc4[i];
  float bet = beta[i];
  float e0 = NEGINF, e1 = NEGINF, e2 = NEGINF;
  if (t >= 1 && t < Tt - 1)                         e0 = alpha[i - Vv]     + s.x + bet;
  if (v >= 1 && v < Vv - 1)                         e1 = alpha[i - 1]      + s.y + bet;
  if (t >= 1 && t < Tt - 1 && v >= 1 && v < Vv - 1) e2 = alpha[i - Vv - 1] + s.z + bet;
  float m = fmaxf(fmaxf(e0, e1), e2);
  float4 o;
  if (m == NEGINF) {
    o = make_float4(0.f, 0.f, 0.f, 0.f);            // norm==-inf -> where() -> 0
  } else {
    float norm = m + logf(expf(e0 - m) + expf(e1 - m) + expf(e2 - m));
    o = make_float4(expf(e0 - norm), expf(e1 - norm), expf(e2 - norm), 0.f);
  }
  out4[i] = o;
}

// ---------------------------------------------------------------------------
extern "C" void kernel_launch(void* const* d_in, const int* in_sizes, int n_in,
                              void* d_out, int out_size, void* d_ws, size_t ws_size,
                              hipStream_t stream) {
  (void)in_sizes; (void)n_in; (void)out_size; (void)ws_size;
  const float* logits = (const float*)d_in[0];
  const size_t N = (size_t)Bb * Tt * Vv * Cc;
  float* scores   = (float*)d_out;                 // output 0
  float* expected = scores + N;                    // output 1
  float* scalar   = scores + 2 * N;                // output 2 (alpha[0,-1,-1])
  float* alpha    = (float*)d_ws;                  // 16 MB
  float* beta     = alpha + (size_t)Bb * Tt * Vv;  // 16 MB

  const int n4 = Bb * Tt * Vv;                     // 4,194,304 float4 cells
  k_logsoftmax<<<n4 / 256, 256, 0, stream>>>((const float4*)logits, (float4*)scores, n4);
  k_dp<<<2 * Bb, 32, 0, stream>>>(scores, alpha, beta, scalar);
  k_expected<<<n4 / 256, 256, 0, stream>>>((const float4*)scores, alpha, beta, (float4*)expected);
}